// EnhancedSAGEModel_5480378270226
// MI455X (gfx1250) — compile-verified
//
#include <hip/hip_runtime.h>

typedef float v2f __attribute__((ext_vector_type(2)));
typedef float v4f __attribute__((ext_vector_type(4)));
typedef float v8f __attribute__((ext_vector_type(8)));
typedef int   v4i __attribute__((ext_vector_type(4)));

#define NNODES 50000
#define NEDGES 800000
#define DIN    256
#define HDIM   256
#define NLAYERS 4
#define BN_EPS 1e-5f

#define BM 128
#define BK 32
#define PITCH (BK + 4)   // 36 floats: 16B-aligned rows, bank-conflict-free frag reads
#define BNROWS 128

#define AS1 __attribute__((address_space(1)))
#define AS3 __attribute__((address_space(3)))

#if defined(__HIP_DEVICE_COMPILE__) && __has_builtin(__builtin_amdgcn_global_load_async_to_lds_b128)
#define HAS_ASYNC 1
#else
#define HAS_ASYNC 0
#endif

static __device__ __forceinline__ AS3 v4i* lds_cast4(const void* p) {
    // generic LDS pointer: low 32 bits are the LDS offset (aperture rules, ISA 10.2)
    return (AS3 v4i*)(uint32_t)(uintptr_t)p;
}
static __device__ __forceinline__ AS1 v4i* gbl_cast4(const void* p) {
    return (AS1 v4i*)(uintptr_t)p;
}

// ---------------- fill (vectorized) ----------------
__global__ void fill_f4(float* __restrict__ p, long n4, float v) {
    long i = (long)blockIdx.x * blockDim.x + threadIdx.x;
    long stride = (long)gridDim.x * blockDim.x;
    v4f val = {v, v, v, v};
    for (; i < n4; i += stride) ((v4f*)p)[i] = val;
}

// ---------------- in-degree ----------------
__global__ void degree_kernel(const int* __restrict__ dst, float* __restrict__ deg, int E) {
    int i = blockIdx.x * blockDim.x + threadIdx.x;
    if (i < E) atomicAdd(&deg[dst[i]], 1.0f);
}

// ---------------- edge scatter-add: agg[dst] += h[src] ----------------
// 4 edges per 256-thread block, 64 threads per edge, float4 per thread (HDIM=256).
__global__ __launch_bounds__(256)
void scatter_add_kernel(const float* __restrict__ h, const int* __restrict__ src,
                        const int* __restrict__ dst, float* __restrict__ agg, int E) {
    int e = blockIdx.x * 4 + (threadIdx.x >> 6);
    if (e >= E) return;
    int c = (threadIdx.x & 63) << 2;
    int s = __builtin_amdgcn_readfirstlane(src[e]);   // wave-uniform -> SGPR
    int d = __builtin_amdgcn_readfirstlane(dst[e]);
    v4f v = *(const v4f*)(h + (size_t)s * HDIM + c);
    float* a = agg + (size_t)d * HDIM + c;
    atomicAdd(a + 0, v.x);
    atomicAdd(a + 1, v.y);
    atomicAdd(a + 2, v.z);
    atomicAdd(a + 3, v.w);
}

// ---------------- mean fold: agg[m][:] *= 1/max(deg[m],1) ----------------
__global__ __launch_bounds__(256)
void scale_rows_kernel(float* __restrict__ agg, const float* __restrict__ deg, int M) {
    long total = (long)M * (HDIM / 4);
    long stride = (long)gridDim.x * blockDim.x;
    for (long i = (long)blockIdx.x * blockDim.x + threadIdx.x; i < total; i += stride) {
        int row = (int)(i >> 6);                      // HDIM/4 == 64 float4 per row
        float d = deg[row];
        float rs = 1.0f / (d > 1.0f ? d : 1.0f);
        v4f v = ((const v4f*)agg)[i];
        v.x *= rs; v.y *= rs; v.z *= rs; v.w *= rs;
        ((v4f*)agg)[i] = v;
    }
}

// ---------------- WMMA fp32 GEMM: out = A(MxK) * W(NxK)^T ----------------
// RELU: relu final. ACC: out += result. ADDMODE: 0 none, 1 +add, 2 +relu(add*bnscale+bnshift).
// Block: 256 thr = 8 waves; tile 128(M) x 16(N); K in LDS chunks of 32, double-buffered.
// Full tiles stage A/B via GLOBAL_LOAD_ASYNC_TO_LDS (ASYNCcnt), boundary tiles take a
// guarded single-buffer path.
template<int RELU, int ACC, int ADDMODE>
__global__ __launch_bounds__(256)
void gemm_wmma(const float* __restrict__ A, const float* __restrict__ W,
               const float* __restrict__ bias, const float* __restrict__ add,
               const float* __restrict__ bnscale, const float* __restrict__ bnshift,
               float* __restrict__ out, int M, int K, int Ncols) {
    __shared__ float lA[2][BM * PITCH];
    __shared__ float lB[2][16 * PITCH];
    const int tid    = threadIdx.x;
    const int wave   = tid >> 5;
    const int lane   = tid & 31;
    const int laneLo = lane & 15;
    const int hi     = lane >> 4;
    const int m0 = blockIdx.x * BM;
    const int n0 = blockIdx.y * 16;

    v8f acc = {};
    const int nc = K / BK;

    // 8 x V_WMMA_F32_16X16X4_F32 per chunk.
    // A frag (16x4): lanes 0-15 = A[m][k..k+1], lanes 16-31 = A[m][k+2..k+3].
    // B frag (4x16): lanes 0-15 = W[n][k..k+1], lanes 16-31 = W[n][k+2..k+3].
    auto compute = [&](int b) {
        const float* aRow = &lA[b][(wave * 16 + laneLo) * PITCH];
        const float* bRow = &lB[b][laneLo * PITCH];
        #pragma unroll
        for (int kk = 0; kk < BK; kk += 4) {
            int c = kk + hi * 2;
            v2f af = *(const v2f*)(aRow + c);
            v2f bf = *(const v2f*)(bRow + c);
            acc = __builtin_amdgcn_wmma_f32_16x16x4_f32(false, af, false, bf,
                                                        (short)0, acc, false, false);
        }
    };

    if (m0 + BM <= M) {           // full tile: pipelined double-buffer
#if HAS_ASYNC
        auto issue = [&](int chunk, int b) {
            int k0 = chunk * BK;
            #pragma unroll
            for (int it = 0; it < 4; ++it) {
                int idx = tid + it * 256;
                int r = idx >> 3, c = (idx & 7) << 2;
                __builtin_amdgcn_global_load_async_to_lds_b128(
                    gbl_cast4(A + (size_t)(m0 + r) * K + k0 + c),
                    lds_cast4(&lA[b][r * PITCH + c]), 0, 0);
            }
            if (tid < 128) {
                int r = tid >> 3, c = (tid & 7) << 2;
                __builtin_amdgcn_global_load_async_to_lds_b128(
                    gbl_cast4(W + (size_t)(n0 + r) * K + k0 + c),
                    lds_cast4(&lB[b][r * PITCH + c]), 0, 0);
            }
        };
        issue(0, 0);
        for (int i = 0; i < nc; ++i) {
            int cur = i & 1;
            asm volatile("s_wait_asynccnt 0x0" ::: "memory");
            __syncthreads();                       // buf[cur] ready; all past compute(i-1)
            if (i + 1 < nc) issue(i + 1, cur ^ 1); // DMA next chunk during compute
            compute(cur);
        }
#else
        v4f ra[4]; v4f rb = {};
        auto gload = [&](int chunk) {
            int k0 = chunk * BK;
            #pragma unroll
            for (int it = 0; it < 4; ++it) {
                int idx = tid + it * 256;
                int r = idx >> 3, c = (idx & 7) << 2;
                ra[it] = *(const v4f*)(A + (size_t)(m0 + r) * K + k0 + c);
            }
            if (tid < 128) {
                int r = tid >> 3, c = (tid & 7) << 2;
                rb = *(const v4f*)(W + (size_t)(n0 + r) * K + k0 + c);
            }
        };
        auto commit = [&](int b) {
            #pragma unroll
            for (int it = 0; it < 4; ++it) {
                int idx = tid + it * 256;
                int r = idx >> 3, c = (idx & 7) << 2;
                *(v4f*)&lA[b][r * PITCH + c] = ra[it];
            }
            if (tid < 128) {
                int r = tid >> 3, c = (tid & 7) << 2;
                *(v4f*)&lB[b][r * PITCH + c] = rb;
            }
        };
        gload(0);
        for (int i = 0; i < nc; ++i) {
            int cur = i & 1;
            commit(cur);
            __syncthreads();
            if (i + 1 < nc) gload(i + 1);          // overlap next global loads w/ compute
            compute(cur);
        }
#endif
    } else {                       // boundary tile: guarded single-buffer path
        for (int k0 = 0; k0 < K; k0 += BK) {
            #pragma unroll
            for (int it = 0; it < 4; ++it) {
                int idx = tid + it * 256;
                int r = idx >> 3, c = (idx & 7) << 2;
                int gr = m0 + r;
                v4f v = {0.f, 0.f, 0.f, 0.f};
                if (gr < M) v = *(const v4f*)(A + (size_t)gr * K + k0 + c);
                *(v4f*)&lA[0][r * PITCH + c] = v;
            }
            if (tid < 128) {
                int r = tid >> 3, c = (tid & 7) << 2;
                *(v4f*)&lB[0][r * PITCH + c] =
                    *(const v4f*)(W + (size_t)(n0 + r) * K + k0 + c);
            }
            __syncthreads();
            compute(0);
            __syncthreads();
        }
    }

    // Epilogue. C/D layout: VGPR vi -> M = vi + 8*hi, N = lane%16.
    const int ncol = n0 + laneLo;
    float bb = bias ? bias[ncol] : 0.0f;
    float sc = 0.f, sh = 0.f;
    if (ADDMODE == 2) { sc = bnscale[ncol]; sh = bnshift[ncol]; }
    #pragma unroll
    for (int vi = 0; vi < 8; ++vi) {
        int m = m0 + wave * 16 + hi * 8 + vi;
        if (m < M) {
            size_t o = (size_t)m * Ncols + ncol;
            float v = acc[vi] + bb;
            if (ADDMODE == 1) v += add[o];
            if (ADDMODE == 2) {
                float t = add[o] * sc + sh;   // BatchNorm affine
                v += (t > 0.f ? t : 0.f);     // + ReLU fused into skip epilogue
            }
            if (ACC) v += out[o];
            if (RELU) v = v > 0.f ? v : 0.f;
            out[o] = v;
        }
    }
}

// ---------------- BatchNorm column stats (sum, sumsq) ----------------
__global__ __launch_bounds__(256)
void bn_stats_kernel(const float* __restrict__ t, float* __restrict__ sums, int M) {
    int c = threadIdx.x;                     // one column per thread -> coalesced rows
    int r0 = blockIdx.x * BNROWS;
    int r1 = r0 + BNROWS; if (r1 > M) r1 = M;
    float s = 0.f, sq = 0.f;
    for (int r = r0; r < r1; ++r) {
        float v = t[(size_t)r * HDIM + c];
        s += v; sq += v * v;
    }
    atomicAdd(&sums[c], s);
    atomicAdd(&sums[HDIM + c], sq);
}

__global__ void bn_finalize_kernel(const float* __restrict__ sums, const float* __restrict__ g,
                                   const float* __restrict__ b, float* __restrict__ scale,
                                   float* __restrict__ shift, int M) {
    int c = threadIdx.x;
    float mu  = sums[c] / (float)M;
    float var = sums[HDIM + c] / (float)M - mu * mu;
    var = var < 0.f ? 0.f : var;
    float sc = g[c] * rsqrtf(var + BN_EPS);
    scale[c] = sc;
    shift[c] = b[c] - mu * sc;
}

// ---------------- final 256->2 projection + log_softmax ----------------
__global__ __launch_bounds__(256)
void out_kernel(const float* __restrict__ h, const float* __restrict__ ow,
                const float* __restrict__ ob, float* __restrict__ out, int M) {
    int node = blockIdx.x * 8 + (threadIdx.x >> 5);  // one wave32 per node
    int lane = threadIdx.x & 31;
    if (node >= M) return;
    const float* hr = h + (size_t)node * HDIM;
    float s0 = 0.f, s1 = 0.f;
    #pragma unroll
    for (int k = lane; k < HDIM; k += 32) {
        float hv = hr[k];
        s0 += hv * ow[k];
        s1 += hv * ow[HDIM + k];
    }
    #pragma unroll
    for (int off = 16; off > 0; off >>= 1) {
        s0 += __shfl_xor(s0, off, 32);
        s1 += __shfl_xor(s1, off, 32);
    }
    if (lane == 0) {
        s0 += ob[0]; s1 += ob[1];
        float mx  = fmaxf(s0, s1);
        float lse = mx + logf(expf(s0 - mx) + expf(s1 - mx));
        out[node * 2 + 0] = s0 - lse;
        out[node * 2 + 1] = s1 - lse;
    }
}

extern "C" void kernel_launch(void* const* d_in, const int* in_sizes, int n_in,
                              void* d_out, int out_size, void* d_ws, size_t ws_size,
                              hipStream_t stream) {
    const float* x       = (const float*)d_in[0];
    const int*   ei      = (const int*)  d_in[1];
    const float* in_w    = (const float*)d_in[2];
    const float* in_b    = (const float*)d_in[3];
    const float* conv_wl = (const float*)d_in[4];
    const float* conv_bl = (const float*)d_in[5];
    const float* conv_wr = (const float*)d_in[6];
    const float* bn_g    = (const float*)d_in[7];
    const float* bn_b    = (const float*)d_in[8];
    const float* skip_w  = (const float*)d_in[9];
    const float* skip_b  = (const float*)d_in[10];
    const float* mlp_w1  = (const float*)d_in[11];
    const float* mlp_b1  = (const float*)d_in[12];
    const float* mlp_w2  = (const float*)d_in[13];
    const float* mlp_b2  = (const float*)d_in[14];
    const float* out_w   = (const float*)d_in[15];
    const float* out_b   = (const float*)d_in[16];
    float* outp = (float*)d_out;

    const int Nn = NNODES, E = NEDGES;

    // Workspace layout (~205 MB)
    float* ws = (float*)d_ws;
    size_t off = 0;
    float* buf0 = ws + off; off += (size_t)Nn * HDIM;       // ping
    float* buf1 = ws + off; off += (size_t)Nn * HDIM;       // pong
    float* buf2 = ws + off; off += (size_t)Nn * 2 * HDIM;   // conv temp / MLP mid
    float* deg  = ws + off; off += Nn;
    float* sums = ws + off; off += 2 * HDIM;
    float* bnsc = ws + off; off += HDIM;
    float* bnsh = ws + off; off += HDIM;

    const int* src = ei;
    const int* dst = ei + E;

    dim3 blk(256);
    dim3 ggrid((Nn + BM - 1) / BM, HDIM / 16);

    // degrees
    fill_f4<<<64, 256, 0, stream>>>(deg, Nn / 4, 0.0f);
    degree_kernel<<<(E + 255) / 256, 256, 0, stream>>>(dst, deg, E);

    // h = relu(x @ in_w^T + in_b)
    gemm_wmma<1, 0, 0><<<ggrid, blk, 0, stream>>>(x, in_w, in_b, nullptr, nullptr, nullptr,
                                                  buf0, Nn, DIN, HDIM);

    float* P = buf0;   // h_prev
    float* G = buf1;   // agg, then new h
    float* T = buf2;   // conv output (pre-BN)
    for (int i = 0; i < NLAYERS; ++i) {
        const size_t wo = (size_t)i * HDIM * HDIM;
        fill_f4<<<2048, 256, 0, stream>>>(G, (long)Nn * HDIM / 4, 0.0f);
        scatter_add_kernel<<<(E + 3) / 4, 256, 0, stream>>>(P, src, dst, G, E);
        scale_rows_kernel<<<2048, 256, 0, stream>>>(G, deg, Nn);     // mean aggregation
        // T = agg_mean @ wl^T + bl
        gemm_wmma<0, 0, 0><<<ggrid, blk, 0, stream>>>(G, conv_wl + wo, conv_bl + i * HDIM,
                                                      nullptr, nullptr, nullptr,
                                                      T, Nn, HDIM, HDIM);
        // T += h_prev @ wr^T
        gemm_wmma<0, 1, 0><<<ggrid, blk, 0, stream>>>(P, conv_wr + wo, nullptr,
                                                      nullptr, nullptr, nullptr,
                                                      T, Nn, HDIM, HDIM);
        // BatchNorm stats -> scale/shift
        fill_f4<<<1, 128, 0, stream>>>(sums, (2 * HDIM) / 4, 0.0f);
        bn_stats_kernel<<<(Nn + BNROWS - 1) / BNROWS, 256, 0, stream>>>(T, sums, Nn);
        bn_finalize_kernel<<<1, HDIM, 0, stream>>>(sums, bn_g + i * HDIM, bn_b + i * HDIM,
                                                   bnsc, bnsh, Nn);
        // G = relu(bn(T)) + h_prev @ skip^T + skip_b
        gemm_wmma<0, 0, 2><<<ggrid, blk, 0, stream>>>(P, skip_w + wo, skip_b + i * HDIM,
                                                      T, bnsc, bnsh,
                                                      G, Nn, HDIM, HDIM);
        float* tmp = P; P = G; G = tmp;   // P now holds new h
    }

    // MLP: mid = relu(h @ w1^T + b1) ; h2 = mid @ w2^T + b2
    dim3 ggrid2((Nn + BM - 1) / BM, (2 * HDIM) / 16);
    gemm_wmma<1, 0, 0><<<ggrid2, blk, 0, stream>>>(P, mlp_w1, mlp_b1, nullptr, nullptr, nullptr,
                                                   buf2, Nn, HDIM, 2 * HDIM);
    gemm_wmma<0, 0, 0><<<ggrid, blk, 0, stream>>>(buf2, mlp_w2, mlp_b2, nullptr, nullptr, nullptr,
                                                  G, Nn, 2 * HDIM, HDIM);

    // logits + log_softmax
    out_kernel<<<(Nn + 7) / 8, 256, 0, stream>>>(G, out_w, out_b, outp, Nn);
}